// SparseGNNLayer_5128190951731
// MI455X (gfx1250) — compile-verified
//
#include <hip/hip_runtime.h>

// SparseGNNLayer for MI455X (gfx1250, wave32).
//
//   Y  = relu([H[src] | Xe] @ M_W^T + M_b)          [E,64]   (edge kernel, WMMA f32)
//   Z  = segment_sum(Y, dst)                        [N,64]   (global_atomic_add_f32 into d_ws)
//   Hn = relu([H | Z] @ U_W^T + U_b)                [N,64]   (node kernel, WMMA f32)
//
// One wave computes TWO 16-row tiles x all 64 output features using
// V_WMMA_F32_16X16X4_F32: 8 accumulator tiles x 32 K-steps = 256 WMMAs/wave-job.
// Each B (weight) fragment loaded once from LDS feeds two WMMAs (2 M-tiles),
// halving ds traffic per FLOP vs the 1-tile version.

typedef __attribute__((ext_vector_type(2))) float v2f;
typedef __attribute__((ext_vector_type(8))) float v8f;

#define WAVES_PER_BLOCK 8
#define BLOCK_THREADS   256

// Stage W [64 x 128] (row-major, torch convention) into LDS as K-pair float2:
//   wlds[kp*64 + n] = { W[n][2kp], W[n][2kp+1] }   (i.e. B[k][n] pairs)
__device__ __forceinline__ void stage_weights(const float* __restrict__ W, v2f* wlds) {
    for (int idx = threadIdx.x; idx < 64 * 64; idx += BLOCK_THREADS) {
        const int kp = idx >> 6;
        const int n  = idx & 63;
        const float* p = W + n * 128 + (kp << 1);
        v2f w;
        w.x = p[0];
        w.y = p[1];
        wlds[idx] = w;
    }
}

// Two 16x128 A tiles against the shared 128x64 B image in LDS.
// A-fragment layout (ISA 7.12.2, 32-bit A 16x4): lanes 0-15 -> K = {k0, k0+1},
// lanes 16-31 -> K = {k0+2, k0+3}, row M = lane&15.
__device__ __forceinline__ void mma_tile2_128x64(const float* __restrict__ pH0,
                                                 const float* __restrict__ pX0,
                                                 const float* __restrict__ pH1,
                                                 const float* __restrict__ pX1,
                                                 const v2f* __restrict__ wlds,
                                                 int lane, v8f acc0[4], v8f acc1[4]) {
    const int hi = lane >> 4;       // 0: K-pair {0,1}, 1: K-pair {2,3} of the step
    const int lo = lane & 15;
    #pragma unroll
    for (int t = 0; t < 4; ++t) { acc0[t] = (v8f){}; acc1[t] = (v8f){}; }

    #pragma unroll 4
    for (int k0 = 0; k0 < 64; k0 += 4) {
        v2f a0 = *(const v2f*)(pH0 + k0 + 2 * hi);
        v2f a1 = *(const v2f*)(pH1 + k0 + 2 * hi);
        const int widx = ((k0 >> 1) + hi) * 64 + lo;
        #pragma unroll
        for (int t = 0; t < 4; ++t) {
            v2f b = wlds[widx + t * 16];
            acc0[t] = __builtin_amdgcn_wmma_f32_16x16x4_f32(
                false, a0, false, b, (short)0, acc0[t], false, false);
            acc1[t] = __builtin_amdgcn_wmma_f32_16x16x4_f32(
                false, a1, false, b, (short)0, acc1[t], false, false);
        }
    }
    #pragma unroll 4
    for (int k0 = 64; k0 < 128; k0 += 4) {
        v2f a0 = *(const v2f*)(pX0 + (k0 - 64) + 2 * hi);
        v2f a1 = *(const v2f*)(pX1 + (k0 - 64) + 2 * hi);
        const int widx = ((k0 >> 1) + hi) * 64 + lo;
        #pragma unroll
        for (int t = 0; t < 4; ++t) {
            v2f b = wlds[widx + t * 16];
            acc0[t] = __builtin_amdgcn_wmma_f32_16x16x4_f32(
                false, a0, false, b, (short)0, acc0[t], false, false);
            acc1[t] = __builtin_amdgcn_wmma_f32_16x16x4_f32(
                false, a1, false, b, (short)0, acc1[t], false, false);
        }
    }
}

// Bias + ReLU + scatter-add one 16x64 D tile.
// D layout: VGPR r holds M = r + 8*hi, N = lo + 16*t.
__device__ __forceinline__ void scatter_tile(const v8f acc[4], const float bias[4],
                                             const int* __restrict__ id1, int e_base,
                                             float* __restrict__ Z, int hi, int lo) {
    #pragma unroll
    for (int r = 0; r < 8; ++r) {
        const int em  = e_base + r + 8 * hi;
        const int dst = id1[em];
        float* zrow = Z + (size_t)dst * 64;
        #pragma unroll
        for (int t = 0; t < 4; ++t) {
            float y = acc[t][r] + bias[t];
            y = y > 0.0f ? y : 0.0f;
            unsafeAtomicAdd(zrow + t * 16 + lo, y);   // global_atomic_add_f32 (no rtn)
        }
    }
}

__global__ __launch_bounds__(BLOCK_THREADS)
void gnn_edge_msg_kernel(const float* __restrict__ H,
                         const float* __restrict__ Xe,
                         const int*   __restrict__ id0,   // source nodes [E]
                         const int*   __restrict__ id1,   // dest nodes   [E]
                         const float* __restrict__ MW,    // [64,128]
                         const float* __restrict__ Mb,    // [64]
                         float*       __restrict__ Z,     // [N,64] accumulator
                         int ntiles) {
    __shared__ v2f wlds[64 * 64];   // 32 KB weight image
    stage_weights(MW, wlds);
    __syncthreads();

    const int lane = threadIdx.x & 31;
    const int wave = threadIdx.x >> 5;
    const int job  = blockIdx.x * WAVES_PER_BLOCK + wave;
    const int tile0 = job * 2;
    if (tile0 >= ntiles) return;          // wave-uniform: EXEC all-ones for WMMA
    const bool has1 = (tile0 + 1) < ntiles;

    const int hi = lane >> 4;
    const int lo = lane & 15;

    const int e_base0 = tile0 * 16;
    const int e_base1 = has1 ? e_base0 + 16 : e_base0;   // alias tail -> safe reads

    const int e_row0 = e_base0 + lo;
    const int e_row1 = e_base1 + lo;
    const int src0   = id0[e_row0];
    const int src1   = id0[e_row1];

    v8f acc0[4], acc1[4];
    mma_tile2_128x64(H + (size_t)src0 * 64, Xe + (size_t)e_row0 * 64,
                     H + (size_t)src1 * 64, Xe + (size_t)e_row1 * 64,
                     wlds, lane, acc0, acc1);

    float bias[4];
    #pragma unroll
    for (int t = 0; t < 4; ++t) bias[t] = Mb[t * 16 + lo];

    scatter_tile(acc0, bias, id1, e_base0, Z, hi, lo);
    if (has1) scatter_tile(acc1, bias, id1, e_base1, Z, hi, lo);
}

__global__ __launch_bounds__(BLOCK_THREADS)
void gnn_node_update_kernel(const float* __restrict__ H,
                            const float* __restrict__ Z,
                            const float* __restrict__ UW,   // [64,128]
                            const float* __restrict__ Ub,   // [64]
                            float*       __restrict__ out,  // [N,64]
                            int ntiles) {
    __shared__ v2f wlds[64 * 64];
    stage_weights(UW, wlds);
    __syncthreads();

    const int lane = threadIdx.x & 31;
    const int wave = threadIdx.x >> 5;
    const int job  = blockIdx.x * WAVES_PER_BLOCK + wave;
    const int tile0 = job * 2;
    if (tile0 >= ntiles) return;
    const bool has1 = (tile0 + 1) < ntiles;

    const int hi = lane >> 4;
    const int lo = lane & 15;

    const int n_base0 = tile0 * 16;
    const int n_base1 = has1 ? n_base0 + 16 : n_base0;

    const int row0 = n_base0 + lo;
    const int row1 = n_base1 + lo;

    v8f acc0[4], acc1[4];
    mma_tile2_128x64(H + (size_t)row0 * 64, Z + (size_t)row0 * 64,
                     H + (size_t)row1 * 64, Z + (size_t)row1 * 64,
                     wlds, lane, acc0, acc1);

    float bias[4];
    #pragma unroll
    for (int t = 0; t < 4; ++t) bias[t] = Ub[t * 16 + lo];

    #pragma unroll
    for (int r = 0; r < 8; ++r) {
        const int node0 = n_base0 + r + 8 * hi;
        float* orow = out + (size_t)node0 * 64;
        #pragma unroll
        for (int t = 0; t < 4; ++t) {
            float y = acc0[t][r] + bias[t];
            orow[t * 16 + lo] = y > 0.0f ? y : 0.0f;
        }
    }
    if (has1) {
        #pragma unroll
        for (int r = 0; r < 8; ++r) {
            const int node1 = n_base1 + r + 8 * hi;
            float* orow = out + (size_t)node1 * 64;
            #pragma unroll
            for (int t = 0; t < 4; ++t) {
                float y = acc1[t][r] + bias[t];
                orow[t * 16 + lo] = y > 0.0f ? y : 0.0f;
            }
        }
    }
}

__global__ void zero_f4_kernel(float4* __restrict__ p, int n4) {
    const int i = blockIdx.x * blockDim.x + threadIdx.x;
    if (i < n4) p[i] = make_float4(0.f, 0.f, 0.f, 0.f);
}

extern "C" void kernel_launch(void* const* d_in, const int* in_sizes, int n_in,
                              void* d_out, int out_size, void* d_ws, size_t ws_size,
                              hipStream_t stream) {
    const float* H  = (const float*)d_in[0];   // [N,64]
    const float* Xe = (const float*)d_in[1];   // [E,64]
    const int*   id = (const int*)  d_in[2];   // [2,E]
    const float* MW = (const float*)d_in[3];   // [64,128]
    const float* Mb = (const float*)d_in[4];   // [64]
    const float* UW = (const float*)d_in[5];   // [64,128]
    const float* Ub = (const float*)d_in[6];   // [64]

    const int N = in_sizes[0] / 64;
    const int E = in_sizes[1] / 64;
    const int* id0 = id;
    const int* id1 = id + E;

    float* Z = (float*)d_ws;                   // [N,64] scatter accumulator

    // Zero Z (harness poisons d_ws; we need clean zeros every call).
    const int n4 = (N * 64) / 4;
    zero_f4_kernel<<<(n4 + BLOCK_THREADS - 1) / BLOCK_THREADS, BLOCK_THREADS, 0, stream>>>(
        (float4*)Z, n4);

    // Edge message + scatter: 2 tiles of 16 edges per wave.
    const int etiles = E / 16;                 // E is a multiple of 16 here
    const int ejobs  = (etiles + 1) / 2;
    gnn_edge_msg_kernel<<<(ejobs + WAVES_PER_BLOCK - 1) / WAVES_PER_BLOCK,
                          BLOCK_THREADS, 0, stream>>>(H, Xe, id0, id1, MW, Mb, Z, etiles);

    // Node update: 2 tiles of 16 nodes per wave.
    const int ntiles = N / 16;                 // N is a multiple of 16 here
    const int njobs  = (ntiles + 1) / 2;
    gnn_node_update_kernel<<<(njobs + WAVES_PER_BLOCK - 1) / WAVES_PER_BLOCK,
                             BLOCK_THREADS, 0, stream>>>(H, Z, UW, Ub, (float*)d_out, ntiles);
}